// Grace_50070728737442
// MI455X (gfx1250) — compile-verified
//
#include <hip/hip_runtime.h>
#include <hip/hip_bf16.h>

// ---------------------------------------------------------------------------
// GCN encoder x2 + projection heads for MI455X (gfx1250, wave32, WMMA).
// GEMMs use V_WMMA_F32_16X16X4_F32 (fp32 in/out, matches reference numerics).
// Scatter-aggregate uses global f32 atomics (operands L2-resident: 192MB L2).
// ---------------------------------------------------------------------------

typedef __attribute__((ext_vector_type(2))) float v2f;
typedef __attribute__((ext_vector_type(8))) float v8f;

// ---------------- degree / norm ----------------

__global__ void degree_kernel(const int* __restrict__ src,
                              const int* __restrict__ dst,
                              float* __restrict__ deg_out,
                              float* __restrict__ deg_in, int E) {
  int e = blockIdx.x * blockDim.x + threadIdx.x;
  if (e < E) {
    atomicAdd(&deg_out[src[e]], 1.0f);
    atomicAdd(&deg_in[dst[e]], 1.0f);
  }
}

__global__ void norm_kernel(float* __restrict__ deg, int n) {
  int i = blockIdx.x * blockDim.x + threadIdx.x;
  if (i < n) deg[i] = rsqrtf(fmaxf(deg[i], 1.0f));
}

// ---------------- row scaling (h = x * norm[row]), float4 ----------------

__global__ void scale_rows_kernel(float* __restrict__ out,
                                  const float* __restrict__ in,
                                  const float* __restrict__ norm,
                                  int n4, int d4) {
  int i = blockIdx.x * blockDim.x + threadIdx.x;
  if (i >= n4) return;
  int row = i / d4;
  float s = norm[row];
  float4 v = reinterpret_cast<const float4*>(in)[i];
  v.x *= s; v.y *= s; v.z *= s; v.w *= s;
  reinterpret_cast<float4*>(out)[i] = v;
}

// ---------------- edge scatter-add: agg[dst] += h[src] ----------------
// One wave32 per edge; float4 gathers, f32 atomic adds (L2-resident).

template <int D>
__global__ void scatter_add_kernel(const float* __restrict__ h,
                                   const int* __restrict__ src,
                                   const int* __restrict__ dst,
                                   float* __restrict__ agg, int E) {
  int warp = (int)((blockIdx.x * (size_t)blockDim.x + threadIdx.x) >> 5);
  int lane = threadIdx.x & 31;
  if (warp >= E) return;
  int s = src[warp];
  int d = dst[warp];
  const float4* hs = reinterpret_cast<const float4*>(h + (size_t)s * D);
  float* ad = agg + (size_t)d * D;
#pragma unroll
  for (int i = 0; i < D / 128; ++i) {
    float4 v = hs[lane + i * 32];
    int c = (lane + i * 32) * 4;
    atomicAdd(ad + c + 0, v.x);
    atomicAdd(ad + c + 1, v.y);
    atomicAdd(ad + c + 2, v.z);
    atomicAdd(ad + c + 3, v.w);
  }
}

// ---------------- WMMA fp32 GEMM + fused epilogue ----------------
// C[M,NC] = act( (A[M,K] @ W[K,NC]) * rowscale[M] + bias[NC] )
// Block: 256 threads = 8 waves; block tile 64(M) x 64(N).
// Wave tile: 32(M) x 16(N) = two 16x16 accumulators sharing one B fragment
// -> 1 global B load + 1 LDS A load per WMMA, 2 independent WMMA chains.
// K and NC are compile-time so all B loads / C stores use base+imm addressing.
// A tile staged in padded LDS (stride K+4 -> conflict-free fragment reads).
// ACT: 0 = none, 1 = relu, 2 = elu(alpha=1)

template <int K, int NC, int ACT, bool HAS_SCALE>
__global__ __launch_bounds__(256) void gemm_wmma_kernel(
    const float* __restrict__ A, const float* __restrict__ W,
    const float* __restrict__ bias, const float* __restrict__ rowscale,
    float* __restrict__ C, int M) {
  constexpr int PAD = 4;
  constexpr int LDSROW = K + PAD;
  __shared__ float sA[64 * LDSROW];

  const int tid = threadIdx.x;
  const int blockM = blockIdx.x * 64;
  const int blockN = blockIdx.y * 64;

  // Cooperative staging of the 64 x K A-tile (float4, coalesced).
  constexpr int F4_PER_ROW = K / 4;
  constexpr int TOTAL_F4 = 64 * F4_PER_ROW;
  for (int idx = tid; idx < TOTAL_F4; idx += 256) {
    int row = idx / F4_PER_ROW;
    int c4 = idx % F4_PER_ROW;
    float4 v = make_float4(0.f, 0.f, 0.f, 0.f);
    int gr = blockM + row;
    if (gr < M) v = *reinterpret_cast<const float4*>(A + (size_t)gr * K + c4 * 4);
    *reinterpret_cast<float4*>(&sA[row * LDSROW + c4 * 4]) = v;
  }
  __syncthreads();

  const int wave = tid >> 5;
  const int lane = tid & 31;
  const int wn = wave & 3;        // 4 N-subtiles of 16
  const int wmg = wave >> 2;      // 2 M-groups of 32 rows
  const int half = lane >> 4;     // 0: lanes 0-15, 1: lanes 16-31
  const int lm = lane & 15;
  const int col = blockN + wn * 16 + lm;  // this lane's output column

  v8f acc0 = {};
  v8f acc1 = {};
  // LDS fragment bases (k added as immediate inside the loop).
  const float* sArow0 = &sA[(wmg * 32 + lm) * LDSROW + 2 * half];
  const float* sArow1 = sArow0 + 16 * LDSROW;
  // Global B base: rows (k + 2*half) and (k + 2*half + 1), column `col`.
  const float* Wp = W + (size_t)(2 * half) * NC + col;

#pragma unroll 8
  for (int k = 0; k < K; k += 4) {
    // A fragments (16x4 f32): VGPR0 = K0 (lanes 0-15) / K2 (lanes 16-31)
    v2f a0 = *reinterpret_cast<const v2f*>(sArow0 + k);
    v2f a1 = *reinterpret_cast<const v2f*>(sArow1 + k);
    // B fragment (4x16 f32): VGPR0 = rows 0/2, VGPR1 = rows 1/3; shared by both tiles
    v2f b;
    b[0] = Wp[k * NC];
    b[1] = Wp[k * NC + NC];
    acc0 = __builtin_amdgcn_wmma_f32_16x16x4_f32(false, a0, false, b, (short)0,
                                                 acc0, false, false);
    acc1 = __builtin_amdgcn_wmma_f32_16x16x4_f32(false, a1, false, b, (short)0,
                                                 acc1, false, false);
  }

  // Epilogue: C/D layout => VGPR r holds rows r (lanes 0-15) and r+8 (lanes 16-31)
  const float bval = bias[col];
  const int rbase0 = blockM + wmg * 32 + 8 * half;
  if (blockM + 64 <= M) {
    // Interior block: unguarded, base+immediate stores.
#pragma unroll
    for (int t = 0; t < 2; ++t) {
      const int rbase = rbase0 + t * 16;
      float* Cp = C + (size_t)rbase * NC + col;
      const float* rs = rowscale + rbase;
#pragma unroll
      for (int r = 0; r < 8; ++r) {
        float v = (t == 0) ? acc0[r] : acc1[r];
        if (HAS_SCALE) v *= rs[r];
        v += bval;
        if (ACT == 1) v = fmaxf(v, 0.0f);
        else if (ACT == 2) v = (v > 0.0f) ? v : expm1f(v);
        Cp[r * NC] = v;
      }
    }
  } else {
    // Tail block: per-row guard.
#pragma unroll
    for (int t = 0; t < 2; ++t) {
      const int rbase = rbase0 + t * 16;
#pragma unroll
      for (int r = 0; r < 8; ++r) {
        const int row = rbase + r;
        if (row < M) {
          float v = (t == 0) ? acc0[r] : acc1[r];
          if (HAS_SCALE) v *= rowscale[row];
          v += bval;
          if (ACT == 1) v = fmaxf(v, 0.0f);
          else if (ACT == 2) v = (v > 0.0f) ? v : expm1f(v);
          C[(size_t)row * NC + col] = v;
        }
      }
    }
  }
}

// ---------------------------------------------------------------------------

extern "C" void kernel_launch(void* const* d_in, const int* in_sizes, int n_in,
                              void* d_out, int out_size, void* d_ws, size_t ws_size,
                              hipStream_t stream) {
  const float* feat1 = (const float*)d_in[0];
  const float* feat2 = (const float*)d_in[1];
  const int* ei1 = (const int*)d_in[2];
  const int* ei2 = (const int*)d_in[3];
  const float* W1 = (const float*)d_in[4];
  const float* b1 = (const float*)d_in[5];
  const float* W2 = (const float*)d_in[6];
  const float* b2 = (const float*)d_in[7];
  const float* fc1W = (const float*)d_in[8];
  const float* fc1b = (const float*)d_in[9];
  const float* fc2W = (const float*)d_in[10];
  const float* fc2b = (const float*)d_in[11];

  const int N = in_sizes[0] / 128;  // 50000 nodes

  // Workspace layout: norms (2N) + two N x 256 ping-pong buffers (~103 MB).
  float* ws = (float*)d_ws;
  float* norm_src = ws;
  float* norm_dst = ws + N;
  float* bufA = ws + 2 * (size_t)N;      // dense buffer
  float* bufB = bufA + (size_t)N * 256;  // aggregation buffer

  for (int g = 0; g < 2; ++g) {
    const float* feat = g ? feat2 : feat1;
    const int E = (g ? in_sizes[3] : in_sizes[2]) / 2;
    const int* src = g ? ei2 : ei1;
    const int* dst = src + E;
    float* zout = (float*)d_out + (size_t)g * N * 128;

    // Degrees -> rsqrt norms (shared by both layers).
    hipMemsetAsync(norm_src, 0, 2 * (size_t)N * sizeof(float), stream);
    degree_kernel<<<(E + 255) / 256, 256, 0, stream>>>(src, dst, norm_src, norm_dst, E);
    norm_kernel<<<(2 * N + 255) / 256, 256, 0, stream>>>(norm_src, 2 * N);

    // ---- layer 1: D 128 -> 256 ----
    scale_rows_kernel<<<(N * 128 / 4 + 255) / 256, 256, 0, stream>>>(
        bufA, feat, norm_src, N * 128 / 4, 32);
    hipMemsetAsync(bufB, 0, (size_t)N * 128 * sizeof(float), stream);
    scatter_add_kernel<128><<<((size_t)E * 32 + 255) / 256, 256, 0, stream>>>(
        bufA, src, dst, bufB, E);
    {
      dim3 grid((N + 63) / 64, 256 / 64);
      gemm_wmma_kernel<128, 256, 1, true><<<grid, 256, 0, stream>>>(
          bufB, W1, b1, norm_dst, bufA, N);
    }

    // ---- layer 2: D 256 -> 128 ----
    scale_rows_kernel<<<(N * 256 / 4 + 255) / 256, 256, 0, stream>>>(
        bufA, bufA, norm_src, N * 256 / 4, 64);
    hipMemsetAsync(bufB, 0, (size_t)N * 256 * sizeof(float), stream);
    scatter_add_kernel<256><<<((size_t)E * 32 + 255) / 256, 256, 0, stream>>>(
        bufA, src, dst, bufB, E);
    {
      dim3 grid((N + 63) / 64, 128 / 64);
      gemm_wmma_kernel<256, 128, 1, true><<<grid, 256, 0, stream>>>(
          bufB, W2, b2, norm_dst, bufA, N);
    }

    // ---- projection head: elu(h@fc1+b) @ fc2 + b ----
    {
      dim3 grid((N + 63) / 64, 128 / 64);
      gemm_wmma_kernel<128, 128, 2, false><<<grid, 256, 0, stream>>>(
          bufA, fc1W, fc1b, nullptr, bufB, N);
      gemm_wmma_kernel<128, 128, 0, false><<<grid, 256, 0, stream>>>(
          bufB, fc2W, fc2b, nullptr, zout, N);
    }
  }
}